// conv3d_35802847379859
// MI455X (gfx1250) — compile-verified
//
#include <hip/hip_runtime.h>

typedef __attribute__((ext_vector_type(2))) float v2f;
typedef __attribute__((ext_vector_type(8))) float v8f;
typedef __attribute__((ext_vector_type(4))) unsigned v4u;
typedef __attribute__((ext_vector_type(4))) int v4i;
typedef __attribute__((ext_vector_type(8))) int v8i;

#define KVOL   27
#define PAIRS  24576
#define CIN    128
#define COUT   128
#define ROWS_PER_WAVE   16
#define WAVES_PER_BLOCK 8
#define ROWS_PER_BLOCK  (ROWS_PER_WAVE * WAVES_PER_BLOCK)   // 128

#if __has_builtin(__builtin_amdgcn_tensor_load_to_lds) && \
    __has_builtin(__builtin_amdgcn_s_wait_tensorcnt)
#define USE_TDM 1
#else
#define USE_TDM 0
#endif

// One block: TDM-stages kernel[k_off] (128x128 f32 = 64KB) into LDS, 8 waves
// each compute a 16-row x 128-col output tile with V_WMMA_F32_16X16X4_F32 and
// scatter-add with global_atomic_add_f32 (omap rows collide across offsets).
__global__ __launch_bounds__(256)
void spconv_wmma_f32_kernel(const float* __restrict__ in_feats,
                            const float* __restrict__ kernel,
                            const int*   __restrict__ imap,
                            const int*   __restrict__ omap,
                            float*       __restrict__ out)
{
    __shared__ float lds_w[CIN * COUT];            // 64 KB, [k][n] layout

    const int k_off = blockIdx.y;                  // 0..26  kernel offset
    const int tile  = blockIdx.x;                  // 0..191 row tile in PAIRS
    const int tid   = threadIdx.x;
    const int wave  = tid >> 5;
    const int lane  = tid & 31;

    // ---- stage weight slice kernel[k_off] into LDS ----
#if USE_TDM
    if (tid < 32) {
        // D# group 0: count=1 | lds_addr | global_addr[56:0] | type=2
        const unsigned long long gaddr =
            (unsigned long long)(const void*)(kernel + (size_t)k_off * CIN * COUT);
        const unsigned lds_off = (unsigned)(unsigned long long)(void*)lds_w;
        v4u g0;
        g0.x = 1u;                                        // count=1, user mode
        g0.y = lds_off;                                   // lds_addr (bytes)
        g0.z = (unsigned)gaddr;                           // global_addr[31:0]
        g0.w = (unsigned)((gaddr >> 32) & 0x1FFFFFFull)   // global_addr[56:32]
               | 0x80000000u;                             // type=2 ("image")
        // D# group 1: one row of 16384 f32 elements (contiguous 64KB tile)
        v8i g1;
        g1[0] = 0x00020000;          // workgroup_mask=0, data_size=2 (4B)
        g1[1] = 0x40000000;          // tensor_dim0[15:0]=0x4000 in bits 63:48
        g1[2] = 0x00010000;          // tensor_dim0[31:16]=0, tensor_dim1 lo=1
        g1[3] = 0x40000000;          // tensor_dim1 hi=0, tile_dim0=0x4000
        g1[4] = 0x00000001;          // tile_dim1=1, tile_dim2=0
        g1[5] = 16384;               // tensor_dim0_stride lo
        g1[6] = 0x40000000;          // dim0_stride hi=0, dim1_stride lo=0x4000
        g1[7] = 0;                   // dim1_stride hi
        v4i z4 = {0, 0, 0, 0};       // groups 2/3 unused (<=2D tensor)
        v8i z8 = {0, 0, 0, 0, 0, 0, 0, 0};
        __builtin_amdgcn_tensor_load_to_lds(g0, g1, z4, z4, z8, 0);
        __builtin_amdgcn_s_wait_tensorcnt(0);
    }
#else
    {
        const float4* wsrc = (const float4*)(kernel + (size_t)k_off * CIN * COUT);
        float4*       wdst = (float4*)lds_w;
        #pragma unroll
        for (int i = 0; i < (CIN * COUT / 4) / 256; ++i)
            wdst[tid + i * 256] = wsrc[tid + i * 256];
    }
#endif
    __syncthreads();

    const int row_base = tile * ROWS_PER_BLOCK + wave * ROWS_PER_WAVE;
    const int* imap_k = imap + k_off * PAIRS + row_base;
    const int* omap_k = omap + k_off * PAIRS + row_base;

    const int half = lane >> 4;                    // 0: K={k0,k0+1}, 1: K={k0+2,k0+3}
    const int l16  = lane & 15;

    // Gather row index for this lane's A fragment (A: row M=l16, ISA layout)
    const int    a_row = imap_k[l16];
    const float* a_ptr = in_feats + (size_t)a_row * CIN + (half << 1);

    v8f acc[8] = {};                               // 8 n-tiles x 16x16 f32 accum

    for (int k0 = 0; k0 < CIN; k0 += 4) {
        // A fragment: 2 consecutive f32 (8B load), lane-half selects K pair
        v2f a = *(const v2f*)(a_ptr + k0);
        const int kk = k0 + (half << 1);
        #pragma unroll
        for (int n = 0; n < 8; ++n) {
            const int col = (n << 4) + l16;
            v2f b;
            b.x = lds_w[kk * COUT + col];          // B: VGPR0 = row kk
            b.y = lds_w[(kk + 1) * COUT + col];    // B: VGPR1 = row kk+1
            acc[n] = __builtin_amdgcn_wmma_f32_16x16x4_f32(
                         /*neg_a=*/false, a,
                         /*neg_b=*/false, b,
                         /*c_mod=*/(short)0, acc[n],
                         /*reuse_a=*/false, /*reuse_b=*/false);
        }
    }

    // ---- scatter-add: D layout VGPR v -> row v + 8*half, col = l16 ----
    int orow[8];
    #pragma unroll
    for (int v = 0; v < 8; ++v)
        orow[v] = omap_k[v + (half << 3)];

    #pragma unroll
    for (int n = 0; n < 8; ++n) {
        #pragma unroll
        for (int v = 0; v < 8; ++v) {
            atomicAdd(out + (size_t)orow[v] * COUT + (n << 4) + l16, acc[n][v]);
        }
    }
}

extern "C" void kernel_launch(void* const* d_in, const int* in_sizes, int n_in,
                              void* d_out, int out_size, void* d_ws, size_t ws_size,
                              hipStream_t stream) {
    const float* in_feats = (const float*)d_in[0];   // [65536,128] f32
    const float* kernel   = (const float*)d_in[1];   // [27,128,128] f32
    const int*   imap     = (const int*)d_in[2];     // [27,24576] i32
    const int*   omap     = (const int*)d_in[3];     // [27,24576] i32
    float*       out      = (float*)d_out;           // [65536,128] f32

    // Output is accumulated with atomics; harness poisons d_out -> zero it.
    (void)hipMemsetAsync(out, 0, (size_t)out_size * sizeof(float), stream);

    dim3 grid(PAIRS / ROWS_PER_BLOCK, KVOL);         // (192, 27)
    spconv_wmma_f32_kernel<<<grid, dim3(256), 0, stream>>>(
        in_feats, kernel, imap, omap, out);
}